// DWT_77378130805389
// MI455X (gfx1250) — compile-verified
//
#include <hip/hip_runtime.h>
#include <stdint.h>

// Problem constants (from reference setup_inputs)
#define BATCH 16
#define CHAN  64
#define HW    256
#define BC    (BATCH*CHAN)          // 1024 images
// Output offsets in floats (flattened return order)
#define OFF_LL3 ((size_t)0)
#define OFF_LH3 ((size_t)1*BC*32*32)
#define OFF_HL3 ((size_t)2*BC*32*32)
#define OFF_HH3 ((size_t)3*BC*32*32)
#define OFF_LH2 ((size_t)4*BC*32*32)
#define OFF_HL2 (OFF_LH2 + (size_t)BC*64*64)
#define OFF_HH2 (OFF_HL2 + (size_t)BC*64*64)
#define OFF_LH1 (OFF_HH2 + (size_t)BC*64*64)
#define OFF_HL1 (OFF_LH1 + (size_t)BC*128*128)
#define OFF_HH1 (OFF_HL1 + (size_t)BC*128*128)

typedef __attribute__((ext_vector_type(4))) float v4f;

union F4 { v4f v; float f[4]; };

// LDS byte address for async-to-LDS ops: generic flat addresses for LDS carry
// the LDS offset in the low 32 bits; VDST of global_load_async_to_lds_* takes
// that 32-bit LDS address.
__device__ __forceinline__ uint32_t lds_addr32(const void* p) {
  return (uint32_t)(uintptr_t)p;
}

__device__ __forceinline__ void nt_store4(float* p, float a, float b, float c,
                                          float d) {
  v4f v = {a, b, c, d};
  __builtin_nontemporal_store(v, (v4f*)p);
}

__launch_bounds__(256, 4)
__global__ void dwt3_haar_fused(const float* __restrict__ x,
                                const float* __restrict__ dec_lo,
                                const float* __restrict__ dec_hi,
                                float* __restrict__ out) {
  __shared__ float s_in[64 * 64];   // 16 KB input tile
  __shared__ float s_ll1[32 * 32];  // 4 KB level-1 LL
  __shared__ float s_ll2[16 * 16];  // 1 KB level-2 LL

  const int tid = threadIdx.x;
  const int blk = blockIdx.x;
  const int bc  = blk >> 4;                 // image index 0..1023
  const int tl  = blk & 15;                 // tile within image (4x4 of 64x64)
  const int ty  = tl >> 2;
  const int tx  = tl & 3;

  // 2-tap analysis filters (scalar broadcast loads)
  const float lo0 = dec_lo[0], lo1 = dec_lo[1];
  const float hi0 = dec_hi[0], hi1 = dec_hi[1];
  // 2x2 separable kernels: outer(row_filter, col_filter)[di][dj]
  const float kll0 = lo0*lo0, kll1 = lo0*lo1, kll2 = lo1*lo0, kll3 = lo1*lo1;
  const float klh0 = hi0*lo0, klh1 = hi0*lo1, klh2 = hi1*lo0, klh3 = hi1*lo1;
  const float khl0 = lo0*hi0, khl1 = lo0*hi1, khl2 = lo1*hi0, khl3 = lo1*hi1;
  const float khh0 = hi0*hi0, khh1 = hi0*hi1, khh2 = hi1*hi0, khh3 = hi1*hi1;

  // One 1x4 output strip from two 1x8 input rows.
  auto dwt4 = [&](const float* t0, const float* t1, float* ll, float* lh,
                  float* hl, float* hh) {
#pragma unroll
    for (int j = 0; j < 4; ++j) {
      const float a = t0[2*j], b = t0[2*j + 1];
      const float c = t1[2*j], d = t1[2*j + 1];
      ll[j] = kll0*a + kll1*b + kll2*c + kll3*d;
      lh[j] = klh0*a + klh1*b + klh2*c + klh3*d;
      hl[j] = khl0*a + khl1*b + khl2*c + khl3*d;
      hh[j] = khh0*a + khh1*b + khh2*c + khh3*d;
    }
  };

  // ---------------- async stage: 64x64 fp32 tile -> LDS -------------------
  // 1024 chunks of 16 B; 256 lanes x 4 iterations, 16 chunks per tile row.
  const float* img = x + (size_t)bc * (HW * (size_t)HW);
  const int row_base = ty * 64;
  const int col_base = tx * 64;
#pragma unroll
  for (int i = 0; i < 4; ++i) {
    const int k  = (i << 8) + tid;          // chunk id
    const int r  = k >> 4;                  // tile row 0..63
    const int c4 = (k & 15) << 2;           // tile col (float index)
    const float*  gp   = img + (size_t)(row_base + r) * HW + (col_base + c4);
    const uint32_t lptr = lds_addr32(&s_in[(r << 6) + c4]);
    asm volatile("global_load_async_to_lds_b128 %0, %1, off"
                 :: "v"(lptr), "v"(gp) : "memory");
  }
  asm volatile("s_wait_asynccnt 0" ::: "memory");
  __syncthreads();

  // ---------------- level 1: 64x64 -> 4 x 32x32 (1x4 strip/thread) --------
  {
    const int oy  = tid >> 3;               // 0..31
    const int ox4 = (tid & 7) << 2;         // 0,4,...,28
    const float* p0 = &s_in[(oy * 2)     * 64 + ox4 * 2];
    const float* p1 = &s_in[(oy * 2 + 1) * 64 + ox4 * 2];
    F4 r0a, r0b, r1a, r1b;
    r0a.v = *(const v4f*)p0;  r0b.v = *(const v4f*)(p0 + 4);
    r1a.v = *(const v4f*)p1;  r1b.v = *(const v4f*)(p1 + 4);
    float t0[8] = {r0a.f[0], r0a.f[1], r0a.f[2], r0a.f[3],
                   r0b.f[0], r0b.f[1], r0b.f[2], r0b.f[3]};
    float t1[8] = {r1a.f[0], r1a.f[1], r1a.f[2], r1a.f[3],
                   r1b.f[0], r1b.f[1], r1b.f[2], r1b.f[3]};
    float ll[4], lh[4], hl[4], hh[4];
    dwt4(t0, t1, ll, lh, hl, hh);
    *(v4f*)&s_ll1[oy * 32 + ox4] = (v4f){ll[0], ll[1], ll[2], ll[3]};
    const size_t ib = (size_t)bc * (128 * 128);
    const size_t o  = (size_t)(ty * 32 + oy) * 128 + (tx * 32 + ox4);
    nt_store4(out + OFF_LH1 + ib + o, lh[0], lh[1], lh[2], lh[3]);
    nt_store4(out + OFF_HL1 + ib + o, hl[0], hl[1], hl[2], hl[3]);
    nt_store4(out + OFF_HH1 + ib + o, hh[0], hh[1], hh[2], hh[3]);
  }
  __syncthreads();

  // ---------------- level 2: 32x32 -> 4 x 16x16 (64 threads) --------------
  if (tid < 64) {
    const int oy  = tid >> 2;               // 0..15
    const int ox4 = (tid & 3) << 2;         // 0,4,8,12
    const float* p0 = &s_ll1[(oy * 2)     * 32 + ox4 * 2];
    const float* p1 = &s_ll1[(oy * 2 + 1) * 32 + ox4 * 2];
    F4 r0a, r0b, r1a, r1b;
    r0a.v = *(const v4f*)p0;  r0b.v = *(const v4f*)(p0 + 4);
    r1a.v = *(const v4f*)p1;  r1b.v = *(const v4f*)(p1 + 4);
    float t0[8] = {r0a.f[0], r0a.f[1], r0a.f[2], r0a.f[3],
                   r0b.f[0], r0b.f[1], r0b.f[2], r0b.f[3]};
    float t1[8] = {r1a.f[0], r1a.f[1], r1a.f[2], r1a.f[3],
                   r1b.f[0], r1b.f[1], r1b.f[2], r1b.f[3]};
    float ll[4], lh[4], hl[4], hh[4];
    dwt4(t0, t1, ll, lh, hl, hh);
    *(v4f*)&s_ll2[oy * 16 + ox4] = (v4f){ll[0], ll[1], ll[2], ll[3]};
    const size_t ib = (size_t)bc * (64 * 64);
    const size_t o  = (size_t)(ty * 16 + oy) * 64 + (tx * 16 + ox4);
    nt_store4(out + OFF_LH2 + ib + o, lh[0], lh[1], lh[2], lh[3]);
    nt_store4(out + OFF_HL2 + ib + o, hl[0], hl[1], hl[2], hl[3]);
    nt_store4(out + OFF_HH2 + ib + o, hh[0], hh[1], hh[2], hh[3]);
  }
  __syncthreads();

  // ---------------- level 3: 16x16 -> 4 x 8x8 (16 threads) ----------------
  if (tid < 16) {
    const int oy  = tid >> 1;               // 0..7
    const int ox4 = (tid & 1) << 2;         // 0,4
    const float* p0 = &s_ll2[(oy * 2)     * 16 + ox4 * 2];
    const float* p1 = &s_ll2[(oy * 2 + 1) * 16 + ox4 * 2];
    F4 r0a, r0b, r1a, r1b;
    r0a.v = *(const v4f*)p0;  r0b.v = *(const v4f*)(p0 + 4);
    r1a.v = *(const v4f*)p1;  r1b.v = *(const v4f*)(p1 + 4);
    float t0[8] = {r0a.f[0], r0a.f[1], r0a.f[2], r0a.f[3],
                   r0b.f[0], r0b.f[1], r0b.f[2], r0b.f[3]};
    float t1[8] = {r1a.f[0], r1a.f[1], r1a.f[2], r1a.f[3],
                   r1b.f[0], r1b.f[1], r1b.f[2], r1b.f[3]};
    float ll[4], lh[4], hl[4], hh[4];
    dwt4(t0, t1, ll, lh, hl, hh);
    const size_t ib = (size_t)bc * (32 * 32);
    const size_t o  = (size_t)(ty * 8 + oy) * 32 + (tx * 8 + ox4);
    nt_store4(out + OFF_LL3 + ib + o, ll[0], ll[1], ll[2], ll[3]);
    nt_store4(out + OFF_LH3 + ib + o, lh[0], lh[1], lh[2], lh[3]);
    nt_store4(out + OFF_HL3 + ib + o, hl[0], hl[1], hl[2], hl[3]);
    nt_store4(out + OFF_HH3 + ib + o, hh[0], hh[1], hh[2], hh[3]);
  }
}

extern "C" void kernel_launch(void* const* d_in, const int* in_sizes, int n_in,
                              void* d_out, int out_size, void* d_ws, size_t ws_size,
                              hipStream_t stream) {
  (void)in_sizes; (void)n_in; (void)d_ws; (void)ws_size; (void)out_size;
  const float* x      = (const float*)d_in[0];
  const float* dec_lo = (const float*)d_in[1];
  const float* dec_hi = (const float*)d_in[2];
  float* out = (float*)d_out;
  // 1024 images x 16 tiles (64x64 input each) = 16384 workgroups of 8 wave32.
  dim3 grid(BC * 16), block(256);
  hipLaunchKernelGGL(dwt3_haar_fused, grid, block, 0, stream, x, dec_lo, dec_hi, out);
}